// LSTMNet_77867757076786
// MI455X (gfx1250) — compile-verified
//
#include <hip/hip_runtime.h>

// Fused 3-layer LSTM (H=6, in=1) over T=512, B=8192, + final FC(6->1).
// Per wave (wave32): 16 batch columns. gates[24x16] = W_aug[24xK] @ state[Kx16]
// via V_WMMA_F32_16X16X4_F32 (M-tiles rows 0-15 / 16-31, K-steps of 4).
// Weights live in VGPRs (A operand, loop invariant); bias folded into C init.

#define T_LEN 512
#define HID 6

typedef __attribute__((ext_vector_type(2))) float v2f;
typedef __attribute__((ext_vector_type(8))) float v8f;

__device__ __forceinline__ float tanh_fast(float x) {
  float y;
  asm("v_tanh_f32 %0, %1" : "=v"(y) : "v"(x));
  return y;
}

__device__ __forceinline__ float sigm(float x) {
  // sigmoid(x) = 0.5*tanh(0.5x) + 0.5
  return __builtin_fmaf(0.5f, tanh_fast(0.5f * x), 0.5f);
}

__device__ __forceinline__ float swap16(float x) {
  // ds_swizzle group-of-32 SWAPX16: xor=0x10, or=0, and=0x1f -> 0x401F
  return __int_as_float(__builtin_amdgcn_ds_swizzle(__float_as_int(x), 0x401F));
}

__device__ __forceinline__ v8f wmma4(v2f a, v2f b, v8f c) {
  return __builtin_amdgcn_wmma_f32_16x16x4_f32(false, a, false, b, (short)0, c,
                                               false, false);
}

// G0: gate rows 0-15 tile, G1: rows 16-31 tile (C/D layout).
// Lower lane n (batch n): G0[r]=row r, swap16(G0[r])=row 8+r, G1[r]=row 16+r.
// Gate order (PyTorch): i=rows 0-5, f=6-11, g=12-17, o=18-23.
// Valid results produced in lanes 0-15; upper lanes compute discarded garbage.
__device__ __forceinline__ void lstm_cell(const v8f G0, const v8f G1,
                                          float c[HID], float hn[HID]) {
  float D0[8];
#pragma unroll
  for (int r = 0; r < 8; ++r) D0[r] = swap16(G0[r]);
  const float ig[HID] = {G0[0], G0[1], G0[2], G0[3], G0[4], G0[5]};
  const float fg[HID] = {G0[6], G0[7], D0[0], D0[1], D0[2], D0[3]};
  const float gg[HID] = {D0[4], D0[5], D0[6], D0[7], G1[0], G1[1]};
  const float og[HID] = {G1[2], G1[3], G1[4], G1[5], G1[6], G1[7]};
#pragma unroll
  for (int j = 0; j < HID; ++j) {
    float si = sigm(ig[j]);
    float sf = sigm(fg[j]);
    float so = sigm(og[j]);
    float tg = tanh_fast(gg[j]);
    c[j] = __builtin_fmaf(sf, c[j], si * tg);
    hn[j] = so * tanh_fast(c[j]);
  }
}

__global__ __launch_bounds__(256) void lstm3_wmma_kernel(
    const float* __restrict__ x,
    const float* __restrict__ Wih0, const float* __restrict__ Whh0,
    const float* __restrict__ bih0, const float* __restrict__ bhh0,
    const float* __restrict__ Wih1, const float* __restrict__ Whh1,
    const float* __restrict__ bih1, const float* __restrict__ bhh1,
    const float* __restrict__ Wih2, const float* __restrict__ Whh2,
    const float* __restrict__ bih2, const float* __restrict__ bhh2,
    const float* __restrict__ fcw, const float* __restrict__ fcb,
    float* __restrict__ out, int Btot) {
  const int lane = threadIdx.x & 31;
  const int wave = (blockIdx.x * blockDim.x + threadIdx.x) >> 5;
  const int up = lane >> 4;   // 0 = lanes 0-15, 1 = lanes 16-31
  const int ml = lane & 15;   // batch column within the 16-wide tile
  const int base = wave * 16;

  // ---- A-operand (weights), loop invariant. f32 16x4 A layout:
  // lane<16: (m=lane, k=s*4+{0,1}); lane>=16: (m=lane-16, k=s*4+{2,3}).
  auto aug0 = [&](int row, int k) -> float {  // layer0: [Whh(6) | Wih(1) | 0]
    if (row >= 24) return 0.f;
    if (k < 6) return Whh0[row * 6 + k];
    if (k == 6) return Wih0[row];
    return 0.f;
  };
  auto augN = [&](const float* Whh, const float* Wih, int row, int k) -> float {
    if (row >= 24) return 0.f;                 // layers1/2: [Whh(6) | Wih(6)]
    if (k < 6) return Whh[row * 6 + k];
    if (k < 12) return Wih[row * 6 + (k - 6)];
    return 0.f;
  };

  v2f A0[2][2], A1[2][3], A2[2][3];
#pragma unroll
  for (int mt = 0; mt < 2; ++mt) {
    const int row = mt * 16 + ml;
#pragma unroll
    for (int s = 0; s < 2; ++s) {
      const int k0 = s * 4 + up * 2;
      A0[mt][s].x = aug0(row, k0);
      A0[mt][s].y = aug0(row, k0 + 1);
    }
#pragma unroll
    for (int s = 0; s < 3; ++s) {
      const int k0 = s * 4 + up * 2;
      A1[mt][s].x = augN(Whh1, Wih1, row, k0);
      A1[mt][s].y = augN(Whh1, Wih1, row, k0 + 1);
      A2[mt][s].x = augN(Whh2, Wih2, row, k0);
      A2[mt][s].y = augN(Whh2, Wih2, row, k0 + 1);
    }
  }

  // ---- Bias folded into accumulator init (C/D layout: vgpr r holds row
  // mt*16 + r for lanes 0-15, row mt*16 + 8 + r for lanes 16-31).
  auto bsum = [&](const float* bi, const float* bh, int row) -> float {
    return (row < 24) ? (bi[row] + bh[row]) : 0.f;
  };
  v8f Cb0[2], Cb1[2], Cb2[2];
#pragma unroll
  for (int mt = 0; mt < 2; ++mt)
#pragma unroll
    for (int r = 0; r < 8; ++r) {
      const int row = mt * 16 + up * 8 + r;
      Cb0[mt][r] = bsum(bih0, bhh0, row);
      Cb1[mt][r] = bsum(bih1, bhh1, row);
      Cb2[mt][r] = bsum(bih2, bhh2, row);
    }

  // ---- State (h replicated in both lane halves; c valid in lanes 0-15).
  float h0[HID], c0[HID], h1[HID], c1[HID], h2[HID], c2[HID];
#pragma unroll
  for (int j = 0; j < HID; ++j) {
    h0[j] = c0[j] = h1[j] = c1[j] = h2[j] = c2[j] = 0.f;
  }

  int bidx = base + ml;
  if (bidx >= Btot) bidx = Btot - 1;  // clamp for tail waves (stores guarded)
  const float* xp = x + (size_t)bidx * T_LEN;

  float hn[HID];
  for (int t0 = 0; t0 < T_LEN; t0 += 4) {
    __builtin_prefetch(xp + t0 + 128, 0, 0);
    const float4 xc = *(const float4*)(xp + t0);
    const float xv[4] = {xc.x, xc.y, xc.z, xc.w};
#pragma unroll
    for (int u = 0; u < 4; ++u) {
      const float xt = xv[u];

      // ===== Layer 0: K = [h0(6), x, 0] (2 K-steps) =====
      // f32 4x16 B layout: lane<16 -> (k=s*4+{0,1}, n=lane);
      //                    lane>=16 -> (k=s*4+{2,3}, n=lane-16).
      v2f bA, bB;
      bA.x = up ? h0[2] : h0[0];
      bA.y = up ? h0[3] : h0[1];
      bB.x = up ? xt : h0[4];
      bB.y = up ? 0.f : h0[5];
      v8f G0 = wmma4(A0[0][1], bB, wmma4(A0[0][0], bA, Cb0[0]));
      v8f G1 = wmma4(A0[1][1], bB, wmma4(A0[1][0], bA, Cb0[1]));
      lstm_cell(G0, G1, c0, hn);
#pragma unroll
      for (int j = 0; j < HID; ++j) {  // broadcast new h to both halves
        const float sv = swap16(hn[j]);
        h0[j] = up ? sv : hn[j];
      }

      // ===== Layer 1: K = [h1(6), h0_new(6)] (3 K-steps) =====
      v2f bC, bD, bE;
      bC.x = up ? h1[2] : h1[0];
      bC.y = up ? h1[3] : h1[1];
      bD.x = up ? h0[0] : h1[4];
      bD.y = up ? h0[1] : h1[5];
      bE.x = up ? h0[4] : h0[2];
      bE.y = up ? h0[5] : h0[3];
      G0 = wmma4(A1[0][2], bE, wmma4(A1[0][1], bD, wmma4(A1[0][0], bC, Cb1[0])));
      G1 = wmma4(A1[1][2], bE, wmma4(A1[1][1], bD, wmma4(A1[1][0], bC, Cb1[1])));
      lstm_cell(G0, G1, c1, hn);
#pragma unroll
      for (int j = 0; j < HID; ++j) {
        const float sv = swap16(hn[j]);
        h1[j] = up ? sv : hn[j];
      }

      // ===== Layer 2: K = [h2(6), h1_new(6)] (3 K-steps) =====
      bC.x = up ? h2[2] : h2[0];
      bC.y = up ? h2[3] : h2[1];
      bD.x = up ? h1[0] : h2[4];
      bD.y = up ? h1[1] : h2[5];
      bE.x = up ? h1[4] : h1[2];
      bE.y = up ? h1[5] : h1[3];
      G0 = wmma4(A2[0][2], bE, wmma4(A2[0][1], bD, wmma4(A2[0][0], bC, Cb2[0])));
      G1 = wmma4(A2[1][2], bE, wmma4(A2[1][1], bD, wmma4(A2[1][0], bC, Cb2[1])));
      lstm_cell(G0, G1, c2, hn);
#pragma unroll
      for (int j = 0; j < HID; ++j) {
        const float sv = swap16(hn[j]);
        h2[j] = up ? sv : hn[j];
      }
    }
  }

  // ---- Final FC: out[b] = fc_w . h2_T + fc_b (one value per batch column)
  if (!up) {
    const int b = base + ml;
    if (b < Btot) {
      float acc = fcb[0];
#pragma unroll
      for (int j = 0; j < HID; ++j) acc = __builtin_fmaf(fcw[j], h2[j], acc);
      out[b] = acc;
    }
  }
}

extern "C" void kernel_launch(void* const* d_in, const int* in_sizes, int n_in,
                              void* d_out, int out_size, void* d_ws,
                              size_t ws_size, hipStream_t stream) {
  const float* x = (const float*)d_in[0];
  const float* Wih0 = (const float*)d_in[1];
  const float* Whh0 = (const float*)d_in[2];
  const float* bih0 = (const float*)d_in[3];
  const float* bhh0 = (const float*)d_in[4];
  const float* Wih1 = (const float*)d_in[5];
  const float* Whh1 = (const float*)d_in[6];
  const float* bih1 = (const float*)d_in[7];
  const float* bhh1 = (const float*)d_in[8];
  const float* Wih2 = (const float*)d_in[9];
  const float* Whh2 = (const float*)d_in[10];
  const float* bih2 = (const float*)d_in[11];
  const float* bhh2 = (const float*)d_in[12];
  const float* fcw = (const float*)d_in[13];
  const float* fcb = (const float*)d_in[14];

  const int Btot = in_sizes[0] / T_LEN;       // x is [B, 512, 1]
  const int waves = (Btot + 15) / 16;         // 16 batch columns per wave
  const int threads = 256;                    // 8 waves per block
  const int blocks = (waves * 32 + threads - 1) / threads;

  lstm3_wmma_kernel<<<blocks, threads, 0, stream>>>(
      x, Wih0, Whh0, bih0, bhh0, Wih1, Whh1, bih1, bhh1, Wih2, Whh2, bih2,
      bhh2, fcw, fcb, (float*)d_out, Btot);
}